// ContextEncodingModule_23424751632803
// MI455X (gfx1250) — compile-verified
//
#include <hip/hip_runtime.h>

// MI455X / gfx1250 fused ContextEncodingModule:
//   k0: zero accumulator workspace
//   k1: fused distance-GEMM (WMMA f32 16x16x4) -> softmax -> aggregation-GEMM (WMMA),
//       x chunks staged to LDS via ASYNC global->LDS copies (ASYNCcnt),
//       VGPR-resident accumulators, atomic flush to (B,K,C) workspace
//   k2: BN/ReLU/sum-over-k -> enc -> attn (sigmoid enc@W_enc) + se_out (sigmoid enc@W_se)
//   k3: bandwidth-bound float4 scale: out = attn * inputs

typedef float v2f __attribute__((ext_vector_type(2)));
typedef float v8f __attribute__((ext_vector_type(8)));
typedef int   v4i __attribute__((ext_vector_type(4)));
typedef v4i __attribute__((address_space(1)))* gv4i_p;   // global 16B granule
typedef v4i __attribute__((address_space(3)))* lv4i_p;   // LDS 16B granule

#define B_   16
#define H_   128
#define W_   128
#define C_   256
#define K_   32
#define CLS_ 21
#define N_   (H_ * W_)        // 16384 points per batch

#define NP        32          // points per chunk staged in LDS
#define XS_STRIDE (C_ + 1)    // 257 pad: column reads dodge the 64-bank conflict
#define DS_STRIDE (K_ + 1)    // 33
#define GPB       32          // workgroups per batch
#define CPW       16          // chunks per workgroup  (GPB*CPW*NP == N_)
#define THREADS   128         // 4 waves (wave32)

#if __has_builtin(__builtin_amdgcn_global_load_async_to_lds_b128) && \
    __has_builtin(__builtin_amdgcn_s_wait_asynccnt)
#define USE_ASYNC_LDS 1
#else
#define USE_ASYNC_LDS 0
#endif

__device__ __forceinline__ float sigmoidf_(float x) { return 1.f / (1.f + __expf(-x)); }

__global__ void ce_zero_kernel(float* __restrict__ p, int n) {
    for (int i = blockIdx.x * blockDim.x + threadIdx.x; i < n; i += gridDim.x * blockDim.x)
        p[i] = 0.f;
}

__global__ __launch_bounds__(THREADS)
void ce_encode_kernel(const float* __restrict__ x,    // (B, N, C)
                      const float* __restrict__ cw,   // (K, C)
                      const float* __restrict__ sm,   // (K,)
                      float* __restrict__ agg,        // ws (B, K, C)  accum
                      float* __restrict__ wsum)       // ws (B, K)     accum
{
    __shared__ __align__(16) float xs[NP][XS_STRIDE]; // x chunk
    __shared__ float ds[NP][DS_STRIDE];               // dot tile, then softmax weights
    __shared__ float x2s[NP];
    __shared__ float c2s[K_];
    __shared__ float sms[K_];
    __shared__ float wsl[K_];

    const int tid  = threadIdx.x;
    const int lane = tid & 31;
    const int wv   = tid >> 5;            // wave id 0..3
    const int b    = blockIdx.x / GPB;
    const int g    = blockIdx.x % GPB;

    // ||c_k||^2, smoothing, wsum init (codewords stay L2-resident: 32 KB)
    if (tid < K_) {
        float s = 0.f;
        const float* cr = cw + tid * C_;
        for (int c = 0; c < C_; ++c) s += cr[c] * cr[c];
        c2s[tid] = s;
        sms[tid] = sm[tid];
        wsl[tid] = 0.f;
    }

    // Persistent GEMM2 accumulators: wave covers (cw-tile = wv>>1, 8 ch-tiles)
    v8f acc2[8];
    for (int t = 0; t < 8; ++t) acc2[t] = (v8f){0.f, 0.f, 0.f, 0.f, 0.f, 0.f, 0.f, 0.f};
    const int cwt2   = wv >> 1;
    const int chbase = (wv & 1) * 8;
    const int l16    = lane & 15;
    const int kh     = (lane >> 4) * 2;   // A/B lane-half K offset (ISA 16x4 f32 layout)

    __syncthreads();

    for (int ci = 0; ci < CPW; ++ci) {
        const int n0 = (g * CPW + ci) * NP;
        const float* xsrc = x + ((size_t)b * N_ + n0) * (size_t)C_;

        // ---- stage x chunk (NP*C = 8192 floats, 16B granules within a row)
#if USE_ASYNC_LDS
        for (int i = tid; i < NP * C_ / 4; i += THREADS) {
            int p = (i * 4) / C_, c = (i * 4) % C_;
            __builtin_amdgcn_global_load_async_to_lds_b128(
                (gv4i_p)(xsrc + i * 4),
                (lv4i_p)&xs[p][c],
                /*offset=*/0, /*cpol=*/0);
        }
        if (ci + 1 < CPW) __builtin_prefetch(xsrc + (size_t)NP * C_, 0, 0);
        if (tid < NP) x2s[tid] = 0.f;
        __builtin_amdgcn_s_wait_asynccnt(0);
#else
        {
            const float4* src4 = (const float4*)xsrc;
            for (int i = tid; i < NP * C_ / 4; i += THREADS) {
                float4 v = src4[i];
                int p = (i * 4) / C_, c = (i * 4) % C_;
                xs[p][c] = v.x; xs[p][c + 1] = v.y; xs[p][c + 2] = v.z; xs[p][c + 3] = v.w;
            }
            if (ci + 1 < CPW) __builtin_prefetch(xsrc + (size_t)NP * C_, 0, 0);
            if (tid < NP) x2s[tid] = 0.f;
        }
#endif
        __syncthreads();

        // ---- ||x_n||^2: 4 threads per point, 64-channel partials + ds_add_f32
        {
            const int p = tid >> 2, c0 = (tid & 3) * 64;
            float s = 0.f;
            for (int c = 0; c < 64; ++c) { float v = xs[p][c0 + c]; s += v * v; }
            atomicAdd(&x2s[p], s);
        }

        // ---- GEMM1: dot[p,k] = x_p . c_k  (16x16 tile per wave, K-dim = C in steps of 4)
        {
            const int pt   = (wv & 1) * 16;        // point tile base
            const int ct   = (wv >> 1) * 16;       // codeword tile base
            const int prow = pt + l16;
            const float* cwrow = cw + (size_t)(ct + l16) * C_;
            v8f acc = (v8f){0.f, 0.f, 0.f, 0.f, 0.f, 0.f, 0.f, 0.f};
            for (int kc = 0; kc < C_; kc += 4) {
                v2f a, bb;
                a.x  = xs[prow][kc + kh];  a.y  = xs[prow][kc + kh + 1];
                bb.x = cwrow[kc + kh];     bb.y = cwrow[kc + kh + 1];
                acc = __builtin_amdgcn_wmma_f32_16x16x4_f32(false, a, false, bb,
                                                            (short)0, acc, false, false);
            }
            const int poff = pt + (lane >> 4) * 8; // C/D: row r (+8 for upper half)
            const int cwi  = ct + l16;
            for (int r = 0; r < 8; ++r) ds[poff + r][cwi] = acc[r];
        }
        __syncthreads();

        // ---- softmax over K with per-k smoothing (x2 term does not cancel)
        if (tid < NP) {
            const float x2 = x2s[tid];
            float m = -3.4e38f;
            for (int k = 0; k < K_; ++k) {
                float v = (x2 + c2s[k] - 2.f * ds[tid][k]) * sms[k];
                ds[tid][k] = v; m = fmaxf(m, v);
            }
            float s = 0.f;
            for (int k = 0; k < K_; ++k) { float e = __expf(ds[tid][k] - m); ds[tid][k] = e; s += e; }
            float inv = 1.f / s;
            for (int k = 0; k < K_; ++k) {
                float wk = ds[tid][k] * inv;
                ds[tid][k] = wk;
                atomicAdd(&wsl[k], wk);       // ds_add_f32
            }
        }
        __syncthreads();

        // ---- GEMM2: acc2 += wgt^T (32 x NP) * x (NP x 256), K-dim = NP in steps of 4
        {
            const int cwi = cwt2 * 16 + l16;
            for (int p0 = 0; p0 < NP; p0 += 4) {
                v2f a;
                a.x = ds[p0 + kh][cwi];
                a.y = ds[p0 + kh + 1][cwi];
                for (int t2 = 0; t2 < 8; ++t2) {
                    const int ch = (chbase + t2) * 16 + l16;
                    v2f bb;
                    bb.x = xs[p0 + kh][ch];
                    bb.y = xs[p0 + kh + 1][ch];
                    acc2[t2] = __builtin_amdgcn_wmma_f32_16x16x4_f32(false, a, false, bb,
                                                                     (short)0, acc2[t2], false, false);
                }
            }
        }
        __syncthreads();  // xs/ds reused next chunk
    }

    // flush persistent accumulators: global_atomic_add_f32 into (B,K,C)
    {
        const int cwi0 = cwt2 * 16 + (lane >> 4) * 8;
        float* aggb = agg + (size_t)b * K_ * C_;
        for (int t2 = 0; t2 < 8; ++t2) {
            const int ch = (chbase + t2) * 16 + l16;
            for (int r = 0; r < 8; ++r)
                atomicAdd(&aggb[(size_t)(cwi0 + r) * C_ + ch], acc2[t2][r]);
        }
    }
    if (tid < K_) atomicAdd(&wsum[b * K_ + tid], wsl[tid]);
}

__global__ __launch_bounds__(C_)
void ce_finalize_kernel(const float* __restrict__ agg, const float* __restrict__ wsum,
                        const float* __restrict__ cw,
                        const float* __restrict__ bn_g, const float* __restrict__ bn_b,
                        const float* __restrict__ bn_m, const float* __restrict__ bn_v,
                        const float* __restrict__ W_enc, const float* __restrict__ b_enc,
                        const float* __restrict__ W_se, const float* __restrict__ b_se,
                        float* __restrict__ attn, float* __restrict__ se_out)
{
    __shared__ float enc[C_];
    const int b = blockIdx.x, c = threadIdx.x;
    float e = 0.f;
    for (int k = 0; k < K_; ++k) {
        float a  = agg[((size_t)b * K_ + k) * C_ + c] - wsum[b * K_ + k] * cw[k * C_ + c];
        float bn = (a - bn_m[k]) * rsqrtf(bn_v[k] + 1e-3f) * bn_g[k] + bn_b[k];
        e += fmaxf(bn, 0.f);
    }
    enc[c] = e;
    __syncthreads();
    float d = b_enc[c];
    for (int j = 0; j < C_; ++j) d += enc[j] * W_enc[j * C_ + c];
    attn[b * C_ + c] = sigmoidf_(d);
    if (c < CLS_) {
        float d2 = b_se[c];
        for (int j = 0; j < C_; ++j) d2 += enc[j] * W_se[j * CLS_ + c];
        se_out[b * CLS_ + c] = sigmoidf_(d2);
    }
}

__global__ __launch_bounds__(256)
void ce_scale_kernel(const float* __restrict__ x, const float* __restrict__ attn,
                     float* __restrict__ out)
{
    const int PER_B4 = H_ * W_ * C_ / 4;       // 1048576 = 2^20
    const int total  = B_ * PER_B4;            // 16777216
    const float4* x4 = (const float4*)x;
    const float4* a4 = (const float4*)attn;
    float4* o4 = (float4*)out;
    for (int i = blockIdx.x * blockDim.x + threadIdx.x; i < total;
         i += gridDim.x * blockDim.x) {
        int b  = i >> 20;
        int c4 = i & (C_ / 4 - 1);
        float4 a = a4[b * (C_ / 4) + c4];
        float4 v = x4[i];
        v.x *= a.x; v.y *= a.y; v.z *= a.z; v.w *= a.w;
        o4[i] = v;
    }
}

extern "C" void kernel_launch(void* const* d_in, const int* in_sizes, int n_in,
                              void* d_out, int out_size, void* d_ws, size_t ws_size,
                              hipStream_t stream) {
    (void)in_sizes; (void)n_in; (void)out_size; (void)ws_size;
    const float* x   = (const float*)d_in[0];
    const float* cw  = (const float*)d_in[1];
    const float* sm  = (const float*)d_in[2];
    const float* bng = (const float*)d_in[3];
    const float* bnb = (const float*)d_in[4];
    const float* bnm = (const float*)d_in[5];
    const float* bnv = (const float*)d_in[6];
    const float* We  = (const float*)d_in[7];
    const float* be  = (const float*)d_in[8];
    const float* Ws  = (const float*)d_in[9];
    const float* bs  = (const float*)d_in[10];

    float* agg  = (float*)d_ws;                        // B*K*C   = 131072 floats
    float* wsum = agg + (size_t)B_ * K_ * C_;          // B*K     = 512
    float* attn = wsum + (size_t)B_ * K_;              // B*C     = 4096

    float* out_fm = (float*)d_out;                     // B*H*W*C floats
    float* out_se = out_fm + (size_t)B_ * H_ * W_ * C_;// B*CLASSES

    const int nz = B_ * K_ * C_ + B_ * K_;
    ce_zero_kernel<<<(nz + 255) / 256, 256, 0, stream>>>(agg, nz);
    ce_encode_kernel<<<B_ * GPB, THREADS, 0, stream>>>(x, cw, sm, agg, wsum);
    ce_finalize_kernel<<<B_, C_, 0, stream>>>(agg, wsum, cw, bng, bnb, bnm, bnv,
                                              We, be, Ws, bs, attn, out_se);
    ce_scale_kernel<<<8192, 256, 0, stream>>>(x, attn, out_fm);
}